// DynamicDWConv_1_74371653698148
// MI455X (gfx1250) — compile-verified
//
#include <hip/hip_runtime.h>
#include <math.h>
#include <stdint.h>

#define B_  16
#define C_  64
#define H_  256
#define W_  256
#define TH  16                 // rows per conv tile

typedef __attribute__((ext_vector_type(2))) float v2f;
typedef __attribute__((ext_vector_type(8))) float v8f;

// ---------------------------------------------------------------------------
// Kernel 1: per-(b,c) plane mean.  1024 blocks x 256 threads, float4 streaming.
// ---------------------------------------------------------------------------
__global__ void ddw_mean_kernel(const float* __restrict__ x, float* __restrict__ avg) {
    __shared__ float red[256];
    const int p = blockIdx.x;                       // plane = b*C + c
    const float4* x4 = (const float4*)(x + ((size_t)p << 16));   // 65536 floats/plane
    float s = 0.f;
    #pragma unroll 8
    for (int i = 0; i < 64; ++i) {                  // 256 thr * 64 * 4 = 65536
        float4 v = x4[threadIdx.x + i * 256];
        s += v.x + v.y + v.z + v.w;
    }
    red[threadIdx.x] = s;
    __syncthreads();
    for (int off = 128; off > 0; off >>= 1) {
        if (threadIdx.x < off) red[threadIdx.x] += red[threadIdx.x + off];
        __syncthreads();
    }
    if (threadIdx.x == 0) avg[p] = red[0] * (1.0f / 65536.0f);
}

// ---------------------------------------------------------------------------
// Kernel 2: dyn = avg(16x64) @ w1^T(64x576) + b1, sigmoid -> wk (16x576).
// One wave per 16-wide N tile: 36 blocks x 32 threads, V_WMMA_F32_16X16X4_F32.
// A(16x4): lanes0-15 M=lane, v0=K0,v1=K1; lanes16-31 v0=K2,v1=K3.
// B(4x16): mirrored. C/D: VGPR r, lane l -> M = r + 8*(l>=16), N = l%16.
// ---------------------------------------------------------------------------
__global__ void ddw_gemm_wmma_kernel(const float* __restrict__ avg,
                                     const float* __restrict__ w1,   // (576,64) row-major
                                     const float* __restrict__ b1,   // (576,)
                                     float* __restrict__ wk) {       // (16,576)
    const int n0   = blockIdx.x * 16;       // 36 tiles over 576
    const int lane = threadIdx.x;           // 0..31 (wave32)
    const int half = lane >> 4;             // 0/1
    const int lm   = lane & 15;

    v8f c;
    const float bias = b1[n0 + lm];         // bias along N, same for every row
    #pragma unroll
    for (int r = 0; r < 8; ++r) c[r] = bias;

    #pragma unroll
    for (int k0 = 0; k0 < 64; k0 += 4) {
        const int k = k0 + 2 * half;
        v2f a, b;
        a.x = avg[lm * 64 + k];
        a.y = avg[lm * 64 + k + 1];
        b.x = w1[(n0 + lm) * 64 + k];       // B[k][n] = w1[n][k]
        b.y = w1[(n0 + lm) * 64 + k + 1];
        c = __builtin_amdgcn_wmma_f32_16x16x4_f32(
                /*neg_a=*/false, a, /*neg_b=*/false, b,
                /*c_mod=*/(short)0, c, /*reuse_a=*/false, /*reuse_b=*/false);
    }

    #pragma unroll
    for (int r = 0; r < 8; ++r) {
        const int m = r + 8 * half;
        const float v = 1.0f / (1.0f + __expf(-c[r]));   // sigmoid
        wk[m * 576 + n0 + lm] = v;
    }
}

// ---------------------------------------------------------------------------
// Kernel 3: fused depthwise 3x3 conv + bias + (x-y)*fs*x + y.
// grid (H/TH, B*C), block 256. LDS tile (TH+2) x 256 with row halo (18 KB),
// staged with GLOBAL_LOAD_ASYNC_TO_LDS_B128 (memory -> LDS, no VGPR roundtrip).
// ---------------------------------------------------------------------------
__global__ void ddw_conv_kernel(const float* __restrict__ x,
                                const float* __restrict__ wk,        // (B*C, 9)
                                const float* __restrict__ conv_bias, // (C,)
                                const float* __restrict__ fscale,    // (C,)
                                float* __restrict__ out) {
    __shared__ __align__(16) float smem[(TH + 2) * W_];   // 18 * 256 floats
    const int p  = blockIdx.y;              // plane b*C + c
    const int ch = p & (C_ - 1);
    const int h0 = blockIdx.x * TH;
    const int t  = threadIdx.x;             // 0..255

    // per-plane dynamic 3x3 weights + scalars (uniform -> scalar loads)
    float w9[9];
    #pragma unroll
    for (int j = 0; j < 9; ++j) w9[j] = wk[p * 9 + j];
    const float cb = conv_bias[ch];
    const float fs = fscale[ch];

    const size_t plane = (size_t)p << 16;   // * 65536
    const float4* xp4  = (const float4*)(x + plane);

    // stage rows [h0-1, h0+TH] into LDS, zero-padded vertically.
    // gh is wave-uniform (t>>6 constant within a wave32) -> no divergence.
    const int col4  = t & 63;               // float4 column 0..63
    const int rbase = t >> 6;               // 0..3
    #pragma unroll
    for (int i = 0; i < 5; ++i) {
        const int r = rbase + i * 4;        // 0..17 (18 rows)
        if (r < TH + 2) {
            const int f4idx = r * 64 + col4;
            const int gh = h0 + r - 1;
            if (gh >= 0 && gh < H_) {
                // async DMA: global (b128) -> LDS, tracked by ASYNCcnt
                unsigned long long ga = (unsigned long long)(xp4 + gh * 64 + col4);
                unsigned int       la = (unsigned int)(uintptr_t)(smem + f4idx * 4);
                asm volatile("global_load_async_to_lds_b128 %0, %1, off"
                             :: "v"(la), "v"(ga) : "memory");
            } else {
                ((float4*)smem)[f4idx] = make_float4(0.f, 0.f, 0.f, 0.f);
            }
        }
    }
    asm volatile("s_wait_asynccnt 0x0" ::: "memory");   // this wave's DMAs landed
    __syncthreads();                                    // all waves' tiles visible

    const int w = t;                        // output column, one per lane
    float* outp = out + plane;
    #pragma unroll 4
    for (int r = 1; r <= TH; ++r) {
        const float xc = smem[r * W_ + w];
        float y = cb;
        #pragma unroll
        for (int dy = 0; dy < 3; ++dy) {
            const float* row = &smem[(r - 1 + dy) * W_];
            const float left  = (w > 0)      ? row[w - 1] : 0.f;
            const float mid   =                row[w];
            const float right = (w < W_ - 1) ? row[w + 1] : 0.f;
            y = fmaf(w9[dy * 3 + 0], left,  y);
            y = fmaf(w9[dy * 3 + 1], mid,   y);
            y = fmaf(w9[dy * 3 + 2], right, y);
        }
        outp[(size_t)(h0 + r - 1) * W_ + w] = (xc - y) * fs * xc + y;
    }
}

// ---------------------------------------------------------------------------
extern "C" void kernel_launch(void* const* d_in, const int* in_sizes, int n_in,
                              void* d_out, int out_size, void* d_ws, size_t ws_size,
                              hipStream_t stream) {
    const float* x         = (const float*)d_in[0];
    const float* w1        = (const float*)d_in[1];
    const float* b1        = (const float*)d_in[2];
    const float* fscale    = (const float*)d_in[3];
    const float* conv_bias = (const float*)d_in[4];
    float* out = (float*)d_out;

    float* avg = (float*)d_ws;          // 1024 floats
    float* wk  = avg + 1024;            // 16*576 = 9216 floats

    ddw_mean_kernel<<<B_ * C_, 256, 0, stream>>>(x, avg);
    ddw_gemm_wmma_kernel<<<576 / 16, 32, 0, stream>>>(avg, w1, b1, wk);
    ddw_conv_kernel<<<dim3(H_ / TH, B_ * C_), 256, 0, stream>>>(x, wk, conv_bias, fscale, out);
}